// NMS_5549097746496
// MI455X (gfx1250) — compile-verified
//
#include <hip/hip_runtime.h>

// Fused 8x8x8 MaxPool-argmax + MaxUnpool + threshold + equality-mask.
// raw: [2,3,128,256,256] f32. out: same shape, float 1.0/0.0 mask.
//
// Per element:  mask = (x==0) || (k == block_argmax && x > 0.5)
// Only the per-block argmax (first occurrence on ties) is needed.
//
// Memory-bound: 201 MB in + 201 MB out => ~17 us floor at 23.3 TB/s.
// Data path: HBM -> LDS via global_load_async_to_lds_b128 (ASYNCcnt, NT
// temporal hint: zero reuse, don't thrash the 192 MB L2), one LDS read
// into registers, ds_max_u64 block reduction, non-temporal b128 stores.

typedef float __attribute__((ext_vector_type(4))) v4f;

namespace {
constexpr int Wfull   = 256;
constexpr int HW      = 256 * 256;            // H*W plane stride (floats)
constexpr int NSTRIPS = 2 * 3 * 16 * 32 * 2;  // B*C*(D/8)*(H/8)*(W halves) = 6144
}

__global__ __launch_bounds__(256)
void nms_mask_kernel(const float* __restrict__ raw, float* __restrict__ out)
{
    // Strip = 8 D-planes x 8 H-rows x 128 W = 8192 floats = 16 blocks along W.
    __shared__ unsigned long long keys[16];
    __shared__ __align__(16) float xs[8192];

    const int tid = threadIdx.x;
    const int s   = blockIdx.x;                 // grid is exactly NSTRIPS

    const int whalf = s & 1;
    const int hblk  = (s >> 1) & 31;
    const int dblk  = (s >> 6) & 15;
    const int bc    = s >> 10;                  // b*C + c  (0..5)

    const size_t base = (size_t)(bc * 128 + dblk * 8) * HW
                      + (size_t)hblk * 8 * Wfull
                      + (size_t)whalf * 128;

    // Per-thread fixed coordinates: thread owns one w-quad of one block,
    // at window-row h = rb, sweeping window-plane d = i (i = 0..7).
    const int wth = (tid & 31) * 4;             // w in [0,128), quad-aligned
    const int rb  = tid >> 5;                   // h within window (0..7)
    const int wb  = wth >> 3;                   // block index along W (0..15)
    const int kt  = rb * 8 + (wth & 7);         // h*8 + w_local part of k

    if (tid < 16) keys[tid] = 0ull;

    // ---- Phase 1: async DMA the strip into LDS (gfx1250 ASYNCcnt path) ----
    {
        const unsigned lds0 =
            (unsigned)(unsigned long long)(&xs[0]) + (unsigned)(tid * 16);
        const float* g0 = raw + base + (size_t)rb * Wfull + wth;
        #pragma unroll
        for (int i = 0; i < 8; ++i) {
            const unsigned lds_addr     = lds0 + (unsigned)(i * 4096);
            const unsigned long long ga = (unsigned long long)(g0 + (size_t)i * HW);
            asm volatile("global_load_async_to_lds_b128 %0, %1, off th:TH_LOAD_NT"
                         :: "v"(lds_addr), "v"(ga)
                         : "memory");
        }
        asm volatile("s_wait_asynccnt 0" ::: "memory");
    }
    __syncthreads();

    // ---- Read this thread's 32 elements from LDS once, keep in VGPRs ----
    v4f v[8];
    #pragma unroll
    for (int i = 0; i < 8; ++i)
        v[i] = *(const v4f*)(&xs[i * 1024 + tid * 4]);

    // ---- Phase 2: per-block argmax (first occurrence on ties) ----
    // key = (monotone_u32(x) << 9) | (511 - k)  -> max-key == max value, min k.
    {
        unsigned long long best = 0ull;
        #pragma unroll
        for (int i = 0; i < 8; ++i) {
            const int kb = i * 64 + kt;
            #pragma unroll
            for (int j = 0; j < 4; ++j) {
                const unsigned u  = __float_as_uint(v[i][j]);
                const unsigned ou = u ^ ((unsigned)((int)u >> 31) | 0x80000000u);
                const unsigned long long key =
                    ((unsigned long long)ou << 9) | (unsigned)(511 - (kb + j));
                if (key > best) best = key;
            }
        }
        atomicMax(&keys[wb], best);             // ds_max_u64
    }
    __syncthreads();

    const int amax = 511 - (int)(keys[wb] & 511ull);

    // ---- Phase 3: emit mask from registers, non-temporal b128 stores ----
    float* og = out + base + (size_t)rb * Wfull + wth;
    #pragma unroll
    for (int i = 0; i < 8; ++i) {
        const int kb = i * 64 + kt;
        v4f m;
        #pragma unroll
        for (int j = 0; j < 4; ++j) {
            const float x  = v[i][j];
            const bool hit = ((kb + j) == amax) && (x > 0.5f);
            m[j] = (hit || x == 0.0f) ? 1.0f : 0.0f;
        }
        __builtin_nontemporal_store(m, (v4f*)(og + (size_t)i * HW));
    }
}

extern "C" void kernel_launch(void* const* d_in, const int* in_sizes, int n_in,
                              void* d_out, int out_size, void* d_ws, size_t ws_size,
                              hipStream_t stream)
{
    (void)in_sizes; (void)n_in; (void)out_size; (void)d_ws; (void)ws_size;
    const float* raw = (const float*)d_in[0];
    float*       out = (float*)d_out;
    nms_mask_kernel<<<NSTRIPS, 256, 0, stream>>>(raw, out);
}